// ScaledDotProductAttention_8014408975008
// MI455X (gfx1250) — compile-verified
//
#include <hip/hip_runtime.h>

typedef float v2f __attribute__((ext_vector_type(2)));
typedef float v8f __attribute__((ext_vector_type(8)));

constexpr int SEQ = 2048;
constexpr int DK  = 64;
constexpr int NBH = 4 * 16;             // B*H
constexpr int QBLK = 32;                // queries per workgroup (2 WMMA q-tiles)
constexpr int QT_PER_BH = SEQ / QBLK;   // 64
constexpr int KTILES = SEQ / 16;        // 128
constexpr int VSTR = 18;                // ldsVT row stride (words): even + conflict-friendly

#define WMMA_F32(A, B, C) \
    __builtin_amdgcn_wmma_f32_16x16x4_f32(false, (A), false, (B), (short)0, (C), false, false)

// Compiler-only ordering for intra-wave LDS communication: DS ops from one wave
// execute in order (ISA 7.3), so no hardware fence is needed -- just stop the
// compiler from reordering. (A full __syncthreads() would emit
// s_wait_storecnt_dscnt 0, stalling on global attn stores every iteration.)
#define COMPILER_LDS_FENCE() asm volatile("" ::: "memory")

__global__ __launch_bounds__(32)
void sdpa_wmma_f32_kernel(const float* __restrict__ Q,
                          const float* __restrict__ Km,
                          const float* __restrict__ Vm,
                          const unsigned char* __restrict__ mask,
                          float* __restrict__ ctx_out,
                          float* __restrict__ attn_out)
{
    __shared__ float ldsVT[DK * VSTR];   // V tile TRANSPOSED: [d][k_local], stride 18
    __shared__ float ldsP[2 * 16 * 16];  // P tiles [t][m][n] for A-layout transpose
    __shared__ float ldsInv[QBLK];       // per-row 1/sum

    const int lane = threadIdx.x & 31;
    const int half = lane >> 4;          // 0: lanes 0-15, 1: lanes 16-31
    const int l16  = lane & 15;
    const int koff = half * 2;           // K offset inside a 4-chunk for A/B frags

    const int bh = blockIdx.x >> 6;                     // / QT_PER_BH
    const int q0 = (blockIdx.x & (QT_PER_BH - 1)) << 5; // 32-query block

    const float* Qb = Q  + ((size_t)bh * SEQ + q0) * DK;
    const float* Kb = Km + (size_t)bh * SEQ * DK;
    const float* Vb = Vm + (size_t)bh * SEQ * DK;
    float* ctxB  = ctx_out  + ((size_t)bh * SEQ + q0) * DK;
    float* attnB = attn_out + ((size_t)bh * SEQ + q0) * SEQ;

    // 1/sqrt(64) * log2(e): scores feed exp2 (v_exp_f32 is natively 2^x)
    constexpr float scale = 0.125f * 1.4426950408889634f;

    // ---- Q block as A-fragments for both 16-row tiles, scale folded in ----
    v2f qa0[16], qa1[16];
#pragma unroll
    for (int c = 0; c < 16; ++c) {
        v2f a = *reinterpret_cast<const v2f*>(Qb + (size_t)l16 * DK + 4 * c + koff);
        qa0[c].x = a.x * scale; qa0[c].y = a.y * scale;
        v2f b = *reinterpret_cast<const v2f*>(Qb + (size_t)(16 + l16) * DK + 4 * c + koff);
        qa1[c].x = b.x * scale; qa1[c].y = b.y * scale;
    }

    const v8f vzero = {0.f,0.f,0.f,0.f,0.f,0.f,0.f,0.f};
    v8f acc0[4] = {vzero, vzero, vzero, vzero};   // ctx rows 0-15, d-tiles 0..3
    v8f acc1[4] = {vzero, vzero, vzero, vzero};   // ctx rows 16-31
    float rs0[8] = {0.f,0.f,0.f,0.f,0.f,0.f,0.f,0.f};
    float rs1[8] = {0.f,0.f,0.f,0.f,0.f,0.f,0.f,0.f};

    // ---- per-lane stream pointers, bumped by constant strides each k-tile ----
    const float* kp = Kb + (size_t)l16 * DK + koff;          // K fragment base
    const float* vp = Vb + (size_t)lane * 4;                 // V staging base (float4/lane)
    const unsigned char* mp = mask + ((size_t)bh * SEQ + q0 + 8 * half) * SEQ + l16;
    float* ap = attnB + (size_t)(8 * half) * SEQ + l16;

    // transposed-store base: this lane's float4 covers d = 4*l16..4*l16+3 at k-row `half`
    float* const vst = ldsVT + (l16 * 4 * VSTR + half);
    // B-fragment load base: lane reads (V[kk][d], V[kk+1][d]) contiguously
    const float* const vld = ldsVT + (l16 * VSTR + koff);

    for (int kt = 0; kt < KTILES; ++kt) {
        COMPILER_LDS_FENCE();   // order LDS reuse across iterations (wave-internal)

        // ---- stage V tile (16x64) into LDS transposed; imm-offset b32 stores ----
#pragma unroll
        for (int i = 0; i < 8; ++i) {
            const float4 v = *reinterpret_cast<const float4*>(vp + i * 128);
            vst[i * 2 + 0 * VSTR] = v.x;
            vst[i * 2 + 1 * VSTR] = v.y;
            vst[i * 2 + 2 * VSTR] = v.z;
            vst[i * 2 + 3 * VSTR] = v.w;
        }

        // ---- scores for both q-tiles: each K fragment feeds two WMMAs ----
        v8f sc0 = vzero, sc1 = vzero;
#pragma unroll
        for (int c = 0; c < 16; ++c) {
            const v2f kb = *reinterpret_cast<const v2f*>(kp + 4 * c);
            sc0 = WMMA_F32(qa0[c], kb, sc0);
            sc1 = WMMA_F32(qa1[c], kb, sc1);
        }

        // prefetch next K/V tiles (one byte/lane pulls whole cachelines)
        if (kt + 1 < KTILES) {
            __builtin_prefetch(kp + 16 * DK + (size_t)(lane >> 2) * DK, 0, 0);
            __builtin_prefetch(vp + 16 * DK, 0, 0);
        }

        // ---- e = mask ? 0 : exp2(s'); rowsum; unnormalized attn; stash P ----
        // C/D layout: VGPR r: lanes 0-15 -> M=r, lanes 16-31 -> M=r+8; N=lane%16
#pragma unroll
        for (int r = 0; r < 8; ++r) {
            const unsigned char mb = mp[(size_t)r * SEQ];
            const float ev = mb ? 0.0f : __builtin_amdgcn_exp2f(sc0[r]);
            rs0[r] += ev;
            ap[(size_t)r * SEQ] = ev;                    // rescaled by 1/sum later
            ldsP[(8 * half + r) * 16 + l16] = ev;
        }
#pragma unroll
        for (int r = 0; r < 8; ++r) {
            const unsigned char mb = mp[(size_t)(16 + r) * SEQ];
            const float ev = mb ? 0.0f : __builtin_amdgcn_exp2f(sc1[r]);
            rs1[r] += ev;
            ap[(size_t)(16 + r) * SEQ] = ev;
            ldsP[256 + (8 * half + r) * 16 + l16] = ev;
        }

        COMPILER_LDS_FENCE();

        // ---- P as A-fragments (transpose via LDS) ----
        v2f pa0[4], pa1[4];
#pragma unroll
        for (int c = 0; c < 4; ++c) {
            pa0[c] = *reinterpret_cast<const v2f*>(&ldsP[l16 * 16 + 4 * c + koff]);
            pa1[c] = *reinterpret_cast<const v2f*>(&ldsP[256 + l16 * 16 + 4 * c + koff]);
        }

        // ---- ctx += P @ V : contiguous b64 B-fragments, each feeds two WMMAs ----
#pragma unroll
        for (int c = 0; c < 4; ++c) {
#pragma unroll
            for (int d = 0; d < 4; ++d) {
                const v2f b = *reinterpret_cast<const v2f*>(vld + d * 16 * VSTR + 4 * c);
                acc0[d] = WMMA_F32(pa0[c], b, acc0[d]);
                acc1[d] = WMMA_F32(pa1[c], b, acc1[d]);
            }
        }

        // ---- bump stream pointers by one k-tile ----
        kp += 16 * DK;
        vp += 16 * DK;
        mp += 16;
        ap += 16;
    }

    // ---- reduce rowsums across the 16 lanes sharing a row ----
#pragma unroll
    for (int r = 0; r < 8; ++r) {
        float a = rs0[r], b = rs1[r];
        a += __shfl_xor(a, 1, 32);  b += __shfl_xor(b, 1, 32);
        a += __shfl_xor(a, 2, 32);  b += __shfl_xor(b, 2, 32);
        a += __shfl_xor(a, 4, 32);  b += __shfl_xor(b, 4, 32);
        a += __shfl_xor(a, 8, 32);  b += __shfl_xor(b, 8, 32);
        rs0[r] = a; rs1[r] = b;
    }
    float inv0[8], inv1[8];
#pragma unroll
    for (int r = 0; r < 8; ++r) { inv0[r] = 1.0f / rs0[r]; inv1[r] = 1.0f / rs1[r]; }

    // ---- normalize + store context (both q-tiles) ----
#pragma unroll
    for (int r = 0; r < 8; ++r) {
        const int m = r + 8 * half;
        float* cr0 = ctxB + (size_t)m * DK;
        float* cr1 = ctxB + (size_t)(16 + m) * DK;
#pragma unroll
        for (int d = 0; d < 4; ++d) {
            cr0[d * 16 + l16] = acc0[d][r] * inv0[r];
            cr1[d * 16 + l16] = acc1[d][r] * inv1[r];
        }
    }

    // ---- publish per-row 1/sum, then rescale the attn block (L2-hot) ----
    if (l16 == 0) {
#pragma unroll
        for (int r = 0; r < 8; ++r) {
            ldsInv[half * 8 + r]      = inv0[r];
            ldsInv[16 + half * 8 + r] = inv1[r];
        }
    }
    COMPILER_LDS_FENCE();

    // Row-wise rescale: hoist 1/sum, 16 float4 load/mul/store per row per lane.
    // Re-reads our own just-written rows (L2-hot; same-wave store->load ordered).
#pragma unroll 1
    for (int row = 0; row < QBLK; ++row) {
        const float s = ldsInv[row];
        float4* rp = reinterpret_cast<float4*>(attnB + (size_t)row * SEQ);
#pragma unroll 4
        for (int j = lane; j < SEQ / 4; j += 32) {
            float4 v = rp[j];
            v.x *= s; v.y *= s; v.z *= s; v.w *= s;
            rp[j] = v;
        }
    }
}

extern "C" void kernel_launch(void* const* d_in, const int* in_sizes, int n_in,
                              void* d_out, int out_size, void* d_ws, size_t ws_size,
                              hipStream_t stream) {
    (void)in_sizes; (void)n_in; (void)out_size; (void)d_ws; (void)ws_size;
    const float* Q = (const float*)d_in[0];
    const float* K = (const float*)d_in[1];
    const float* V = (const float*)d_in[2];
    const unsigned char* mask = (const unsigned char*)d_in[3];  // jnp.bool_ = 1 byte

    float* ctx  = (float*)d_out;                                // (B,H,S,D) first
    float* attn = (float*)d_out + (size_t)NBH * SEQ * DK;       // then (B,H,S,S)

    dim3 grid(NBH * QT_PER_BH);   // 4096 workgroups, 1 wave each
    sdpa_wmma_f32_kernel<<<grid, 32, 0, stream>>>(Q, K, V, mask, ctx, attn);
}